// CombinedPeakLoss_71545565217538
// MI455X (gfx1250) — compile-verified
//
#include <hip/hip_runtime.h>

#ifndef __has_builtin
#define __has_builtin(x) 0
#endif

#define DPB   256      // days per block (1 thread = 1 day)
#define ROW   28       // padded LDS row stride in floats (112 B, 16-byte aligned)
#define NEGV  (-1e9f)

typedef int v4i __attribute__((ext_vector_type(4)));
typedef __attribute__((address_space(1))) v4i* g4p;   // global (AS1) int4*
typedef __attribute__((address_space(3))) v4i* l4p;   // LDS (AS3) int4*
typedef __attribute__((address_space(3))) void* las_ptr;

// ---- CDNA5 async global -> LDS copy (ASYNCcnt-tracked) -------------------
__device__ __forceinline__ void async_ld16(const float* g, float* l) {
#if __has_builtin(__builtin_amdgcn_global_load_async_to_lds_b128)
  __builtin_amdgcn_global_load_async_to_lds_b128((g4p)g, (l4p)l, 0, 0);
#else
  unsigned loff = (unsigned)(unsigned long long)(las_ptr)l;   // LDS byte offset
  asm volatile("global_load_async_to_lds_b128 %0, %1, off"
               :: "v"(loff), "v"(g) : "memory");
#endif
}

__device__ __forceinline__ void wait_async0() {
#if __has_builtin(__builtin_amdgcn_s_wait_asynccnt)
  __builtin_amdgcn_s_wait_asynccnt(0);
#else
  asm volatile("s_wait_asynccnt 0x0" ::: "memory");
#endif
}

// ---- main kernel: per-day loss terms, per-block partial sums -------------
__global__ __launch_bounds__(256) void peakloss_partial(
    const float* __restrict__ yP, const float* __restrict__ yT,
    float* __restrict__ partials, int nDays)
{
  __shared__ float sT[DPB * ROW];
  __shared__ float sP[DPB * ROW];
  __shared__ float sRed[8 * 6];

  const int    tid      = threadIdx.x;
  const int    day0     = blockIdx.x * DPB;
  const size_t base     = (size_t)day0 * 24;
  const int    daysHere = min(DPB, nDays - day0);
  const int    nChunks  = daysHere * 6;          // 16B chunks per tensor

  // Coalesced async stage-in: for fixed j, 256 lanes fetch 256 consecutive
  // 16B chunks (4 KB) per tensor; scattered into padded per-day LDS rows.
  #pragma unroll
  for (int j = 0; j < 6; ++j) {
    int c = tid + j * DPB;
    if (c < nChunks) {
      int day = c / 6;
      int k   = c - day * 6;
      int go  = c * 4;               // float offset in global tile
      int lo  = day * ROW + k * 4;   // float offset in LDS (16B aligned)
      async_ld16(yT + base + go, &sT[lo]);
      async_ld16(yP + base + go, &sP[lo]);
    }
  }
  wait_async0();
  __syncthreads();

  float s0 = 0.f, s1 = 0.f, s2 = 0.f, s3 = 0.f, s4 = 0.f, s5 = 0.f;

  if (tid < daysHere) {
    const float* rt = &sT[tid * ROW];
    const float* rp = &sP[tid * ROW];
    float yt[24], yp[24];
    #pragma unroll
    for (int h = 0; h < 24; ++h) { yt[h] = rt[h]; yp[h] = rp[h]; }

    // L_overall + daytime-weighted MSE sums
    #pragma unroll
    for (int h = 0; h < 24; ++h) {
      float d = yp[h] - yt[h];
      float d2 = d * d;
      s0 += d2;
      s1 += (h >= 6 && h <= 20) ? 2.0f * d2 : d2;
    }

    // Soft peaks (softmax over daytime hours, T = 0.1; night attn == 0)
    float tm = yt[6], pm = yp[6];
    #pragma unroll
    for (int h = 7; h <= 20; ++h) { tm = fmaxf(tm, yt[h]); pm = fmaxf(pm, yp[h]); }
    float tse = 0.f, tsv = 0.f, tst = 0.f, pse = 0.f, psv = 0.f, pst = 0.f;
    #pragma unroll
    for (int h = 6; h <= 20; ++h) {
      float et = __expf((yt[h] - tm) * 10.0f);
      float ep = __expf((yp[h] - pm) * 10.0f);
      tse += et; tsv += et * yt[h]; tst += et * (float)h;
      pse += ep; psv += ep * yp[h]; pst += ep * (float)h;
    }
    float tinv = 1.0f / tse, pinv = 1.0f / pse;
    float dv = psv * pinv - tsv * tinv;   // peak value diff
    float dt = pst * pinv - tst * tinv;   // peak time diff
    s2 = dv * dv;
    s3 = dt * dt;

    // Peak shape: hard argmax on daytime truth, +/-2h window
    int idx = 6; float best = yt[6];
    #pragma unroll
    for (int h = 7; h <= 20; ++h) if (yt[h] > best) { best = yt[h]; idx = h; }
    int ws = idx - 2; if (ws < 0)  ws = 0;
    int we = idx + 3; if (we > 24) we = 24;
    float tmax = NEGV, pmax = NEGV;
    #pragma unroll
    for (int h = 0; h < 24; ++h) {
      if (h >= ws && h < we) { tmax = fmaxf(tmax, yt[h]); pmax = fmaxf(pmax, yp[h]); }
    }
    bool  valid = tmax > 1e-6f;
    float tsafe = valid ? tmax : 1.0f;
    float ipn = 1.0f / (pmax + 1e-6f);
    float itn = 1.0f / tsafe;
    float acc = 0.f;
    #pragma unroll
    for (int h = 0; h < 24; ++h) {
      float d = yp[h] * ipn - yt[h] * itn;
      acc += (h >= ws && h < we) ? d * d : 0.0f;
    }
    float mse = acc / (float)(we - ws);
    s4 = valid ? mse : 0.0f;
    s5 = valid ? 1.0f : 0.0f;
  }

  // Deterministic wave32 xor-tree reduction
  #pragma unroll
  for (int off = 16; off > 0; off >>= 1) {
    s0 += __shfl_xor(s0, off, 32);
    s1 += __shfl_xor(s1, off, 32);
    s2 += __shfl_xor(s2, off, 32);
    s3 += __shfl_xor(s3, off, 32);
    s4 += __shfl_xor(s4, off, 32);
    s5 += __shfl_xor(s5, off, 32);
  }
  int lane = tid & 31, wid = tid >> 5;
  if (lane == 0) {
    sRed[wid * 6 + 0] = s0; sRed[wid * 6 + 1] = s1; sRed[wid * 6 + 2] = s2;
    sRed[wid * 6 + 3] = s3; sRed[wid * 6 + 4] = s4; sRed[wid * 6 + 5] = s5;
  }
  __syncthreads();
  if (tid < 6) {
    float t = 0.f;
    #pragma unroll
    for (int w = 0; w < 8; ++w) t += sRed[w * 6 + tid];
    partials[blockIdx.x * 6 + tid] = t;
  }
}

// ---- final deterministic reduction + loss combination --------------------
__global__ __launch_bounds__(256) void peakloss_final(
    const float* __restrict__ partials, float* __restrict__ out,
    int nBlocks, int nDays, long long nElems)
{
  __shared__ double red[256][6];
  const int tid = threadIdx.x;
  double a[6] = {0, 0, 0, 0, 0, 0};
  for (int b = tid; b < nBlocks; b += 256) {
    #pragma unroll
    for (int i = 0; i < 6; ++i) a[i] += (double)partials[b * 6 + i];
  }
  #pragma unroll
  for (int i = 0; i < 6; ++i) red[tid][i] = a[i];
  __syncthreads();
  for (int s = 128; s > 0; s >>= 1) {
    if (tid < s) {
      #pragma unroll
      for (int i = 0; i < 6; ++i) red[tid][i] += red[tid + s][i];
    }
    __syncthreads();
  }
  if (tid == 0) {
    double inv   = 1.0 / (double)nElems;
    double Lo    = red[0][0] * inv;            // L_overall
    double Wm    = red[0][1] * inv;            // weighted MSE
    double Ld    = Wm - Lo;                    // L_daytime
    double invD  = 1.0 / (double)nDays;
    double Lpv   = red[0][2] * invD;           // L_peak_value
    double Lpt   = red[0][3] * invD;           // L_peak_time
    double Lps   = red[0][4] / (red[0][5] + 1e-6); // L_peak_shape
    out[0] = (float)(1.0 * Lo + 2.0 * Lpv + 1.0 * Lpt + 0.5 * Lps + 0.5 * Ld);
  }
}

extern "C" void kernel_launch(void* const* d_in, const int* in_sizes, int n_in,
                              void* d_out, int out_size, void* d_ws, size_t ws_size,
                              hipStream_t stream)
{
  const float* yP = (const float*)d_in[0];   // y_pred
  const float* yT = (const float*)d_in[1];   // y_true
  long long total = (long long)in_sizes[0];  // B*S
  int nDays   = (int)(total / 24);           // B*D
  int nBlocks = (nDays + DPB - 1) / DPB;     // 1920 for the reference shape

  float* partials = (float*)d_ws;            // nBlocks*6 floats, written before read

  peakloss_partial<<<nBlocks, 256, 0, stream>>>(yP, yT, partials, nDays);
  peakloss_final<<<1, 256, 0, stream>>>(partials, (float*)d_out, nBlocks, nDays, total);
}